// GRU_41987600286077
// MI455X (gfx1250) — compile-verified
//
#include <hip/hip_runtime.h>
#include <hip/hip_bf16.h>

#define BATCH 256
#define SEQ   1024
#define DRAW  32
#define HID   128
#define KP0   36   // layer-0 input dim padded (34 -> 36)
#define KR0   34
#define KP1   128
#define WSTR  132  // LDS row stride (floats) -> conflict-free b64 reads

typedef __attribute__((ext_vector_type(2))) float v2f;
typedef __attribute__((ext_vector_type(4))) float v4f;
typedef __attribute__((ext_vector_type(8))) float v8f;

// pointer types matching the async-load builtin's expected signature
typedef int b128_t __attribute__((vector_size(16)));
typedef b128_t __attribute__((address_space(1)))* gptr128;
typedef b128_t __attribute__((address_space(3)))* lptr128;

#if defined(__has_builtin)
#if __has_builtin(__builtin_amdgcn_global_load_async_to_lds_b128) && \
    __has_builtin(__builtin_amdgcn_s_wait_asynccnt)
#define HAVE_ASYNC_LDS 1
#endif
#endif
#ifndef HAVE_ASYNC_LDS
#define HAVE_ASYNC_LDS 0
#endif

__device__ __forceinline__ v8f wmma4(v2f a, v2f b, v8f c) {
    // D = A(16x4 f32) * B(4x16 f32) + C(16x16 f32)
    return __builtin_amdgcn_wmma_f32_16x16x4_f32(false, a, false, b, (short)0, c,
                                                 false, false);
}

__device__ __forceinline__ float sigf(float x) {
    return 1.0f / (1.0f + __expf(-x));
}
__device__ __forceinline__ float tanhfast(float x) {
    float e = __expf(-2.0f * x);
    return (1.0f - e) / (1.0f + e);
}

// 16-byte async copy global -> LDS (ASYNCcnt tracked); sync fallback.
__device__ __forceinline__ void copy16_async(const float* g, float* l) {
#if HAVE_ASYNC_LDS
    __builtin_amdgcn_global_load_async_to_lds_b128((gptr128)g, (lptr128)l, 0, 0);
#else
    *(v4f*)l = *(const v4f*)g;
#endif
}
__device__ __forceinline__ void wait_async0() {
#if HAVE_ASYNC_LDS
    __builtin_amdgcn_s_wait_asynccnt(0);
#endif
}

// ---------------------------------------------------------------------------
// Kernel 1: GRU-D imputation. One wave (32 lanes) per batch row; lane = feature.
// imp[b][s][0:32]=forward-filled features, [32]=mask, [33]=t-delta, [34..35]=0.
// ---------------------------------------------------------------------------
__global__ __launch_bounds__(32) void impute_kernel(
        const float* __restrict__ t, const float* __restrict__ x,
        float* __restrict__ imp) {
    const int b = blockIdx.x;
    const int L = threadIdx.x;

    __shared__ float t_s[SEQ];
    for (int i = L; i < SEQ; i += 32) t_s[i] = t[i];
    __syncthreads();

    const size_t xbase = (size_t)b * SEQ * DRAW;
    const size_t obase = (size_t)b * SEQ * KP0;

    float last = 0.0f, lastT = 0.0f;
    int lastIdx = -1;

    const int PF = 8;                 // software pipeline depth
    float buf[PF];
#pragma unroll
    for (int p = 0; p < PF; ++p) buf[p] = x[xbase + (size_t)p * DRAW + L];

    for (int ib = 0; ib < SEQ; ib += PF) {
#pragma unroll
        for (int p = 0; p < PF; ++p) {
            const int i = ib + p;
            float xv = buf[p];
            if (i + PF < SEQ) buf[p] = x[xbase + (size_t)(i + PF) * DRAW + L];

            const bool nanv = (xv != xv);
            const bool m = (__any((int)nanv) != 0);   // wave32-uniform row mask

            const float ti = t_s[i];
            const float te = (lastIdx >= 0) ? (ti - lastT)
                                            : (i == 0 ? 0.0f : ti - t_s[i - 1]);
            const float outv = m ? (lastIdx >= 0 ? last : 0.0f) : xv;

            imp[obase + (size_t)i * KP0 + L] = outv;
            if (L == 0) {
                imp[obase + (size_t)i * KP0 + 32] = m ? 1.0f : 0.0f;
                imp[obase + (size_t)i * KP0 + 33] = te;
                imp[obase + (size_t)i * KP0 + 34] = 0.0f;
                imp[obase + (size_t)i * KP0 + 35] = 0.0f;
            }
            if (!m) { last = xv; lastIdx = i; lastT = ti; }
        }
    }
}

// issue async prefetch of the 16-row input tile for step s into xbuf
template <int KP>
__device__ __forceinline__ void stage_tile(const float* __restrict__ inp,
                                           float* xbuf, int b0, int s, int tid) {
    constexpr int CPR = KP / 4;          // b128 chunks per row
    constexpr int CH  = 16 * CPR;        // total chunks
    for (int i = tid; i < CH; i += 256) {
        const int r  = i / CPR;
        const int c4 = (i - r * CPR) * 4;
        copy16_async(inp + (size_t)(b0 + r) * SEQ * KP + (size_t)s * KP + c4,
                     xbuf + r * WSTR + c4);
    }
}

// ---------------------------------------------------------------------------
// Kernel 2: one GRU layer, f32 WMMA (16x16x4). Grid = 16 blocks (batch tiles
// of 16 rows) x 8 waves; wave w owns hidden cols [16w,16w+16) for r,z,n.
// Whh LDS-resident (padded stride), Wih fragments in VGPRs, input tile
// double-buffered via async global->LDS prefetch.
// ---------------------------------------------------------------------------
template <int KP, int KREAL, int KSTEPS, bool WRITE_ALL>
__global__ __launch_bounds__(256) void gru_wmma(
        const float* __restrict__ inp,   // [B][SEQ][KP]
        const float* __restrict__ Wih,   // [3H][KREAL]
        const float* __restrict__ bih,   // [3H]
        const float* __restrict__ Whh,   // [3H][H]
        const float* __restrict__ bhh,   // [3H]
        float* __restrict__ hall,        // [B][SEQ][H] (WRITE_ALL)
        float* __restrict__ hlast) {     // [B][H]      (!WRITE_ALL)
    const int tid  = threadIdx.x;
    const int wave = tid >> 5;
    const int lane = tid & 31;
    const int half = lane >> 4;
    const int l16  = lane & 15;
    const int b0   = blockIdx.x * 16;
    const int hc   = wave * 16 + l16;   // hidden column owned by this lane

    extern __shared__ float smem[];
    float* whh_s = smem;                            // 3H x WSTR (384*132)
    float* xb0   = smem + 3 * HID * WSTR;           // 16 x WSTR (buffer 0)
    // buffer 1 = xb0 + 16*WSTR
    float* h_s   = smem + (3 * HID + 32) * WSTR;    // 16 x WSTR

    // stage Whh into LDS with padded stride; zero h
    for (int i = tid; i < 3 * HID * HID; i += 256)
        whh_s[(i >> 7) * WSTR + (i & 127)] = Whh[i];
    for (int i = tid; i < 16 * WSTR; i += 256) h_s[i] = 0.0f;

    // prefetch step-0 input tile into buffer 0
    stage_tile<KP>(inp, xb0, b0, 0, tid);

    // per-lane biases (N = hc is constant across the 8 accumulator elements)
    const float bir = bih[hc],           bhr = bhh[hc];
    const float biz = bih[HID + hc],     bhz = bhh[HID + hc];
    const float bin = bih[2 * HID + hc], bhn = bhh[2 * HID + hc];

    // preload Wih B-fragments: B[k][n] = Wih[gate*H + hc][k]
    v2f wf[3][KSTEPS];
#pragma unroll
    for (int g = 0; g < 3; ++g) {
#pragma unroll
        for (int kk = 0; kk < KSTEPS; ++kk) {
            const int k0 = 4 * kk + 2 * half;
            const size_t row = (size_t)(g * HID + hc) * KREAL;
            v2f w;
            w.x = (k0     < KREAL) ? Wih[row + k0]     : 0.0f;
            w.y = (k0 + 1 < KREAL) ? Wih[row + k0 + 1] : 0.0f;
            wf[g][kk] = w;
        }
    }

    wait_async0();
    __syncthreads();   // Whh, h, buffer 0 ready for all waves

    const v8f vzero = {0.f, 0.f, 0.f, 0.f, 0.f, 0.f, 0.f, 0.f};

    for (int s = 0; s < SEQ; ++s) {
        const int cur = s & 1;
        float* xs = xb0 + cur * (16 * WSTR);

        // overlap next step's input prefetch with this step's compute
        if (s + 1 < SEQ)
            stage_tile<KP>(inp, xb0 + (cur ^ 1) * (16 * WSTR), b0, s + 1, tid);

        v8f aXr = vzero, aXz = vzero, aXn = vzero;
        v8f aHr = vzero, aHz = vzero, aHn = vzero;

        // merged K-loop: 6 independent accumulator chains per iteration
#pragma unroll
        for (int kk = 0; kk < HID / 4; ++kk) {
            v2f ah = *(const v2f*)&h_s[l16 * WSTR + 4 * kk + 2 * half];
            v2f br = *(const v2f*)&whh_s[(0 * HID + hc) * WSTR + 4 * kk + 2 * half];
            v2f bz = *(const v2f*)&whh_s[(1 * HID + hc) * WSTR + 4 * kk + 2 * half];
            v2f bn = *(const v2f*)&whh_s[(2 * HID + hc) * WSTR + 4 * kk + 2 * half];
            aHr = wmma4(ah, br, aHr);
            aHz = wmma4(ah, bz, aHz);
            aHn = wmma4(ah, bn, aHn);
            if (kk < KSTEPS) {
                v2f ax = *(const v2f*)&xs[l16 * WSTR + 4 * kk + 2 * half];
                aXr = wmma4(ax, wf[0][kk], aXr);
                aXz = wmma4(ax, wf[1][kk], aXz);
                aXn = wmma4(ax, wf[2][kk], aXn);
            }
        }
        __syncthreads();   // all h_s / xs reads complete before h_s rewrite

        float hv[8];
#pragma unroll
        for (int i = 0; i < 8; ++i) {
            const int m = i + 8 * half;            // batch row within tile
            const float r = sigf(aXr[i] + bir + aHr[i] + bhr);
            const float z = sigf(aXz[i] + biz + aHz[i] + bhz);
            const float n = tanhfast(aXn[i] + bin + r * (aHn[i] + bhn));
            const float hold = h_s[m * WSTR + hc];
            hv[i] = (1.0f - z) * n + z * hold;
        }
#pragma unroll
        for (int i = 0; i < 8; ++i) {
            const int m = i + 8 * half;
            h_s[m * WSTR + hc] = hv[i];
            if (WRITE_ALL) {
                hall[(size_t)(b0 + m) * SEQ * HID + (size_t)s * HID + hc] = hv[i];
            }
        }
        if (!WRITE_ALL && s == SEQ - 1) {
#pragma unroll
            for (int i = 0; i < 8; ++i) {
                const int m = i + 8 * half;
                hlast[(size_t)(b0 + m) * HID + hc] = hv[i];
            }
        }
        wait_async0();     // own prefetch portion landed in LDS
        __syncthreads();   // publish prefetch + h_s update to all waves
    }
}

// ---------------------------------------------------------------------------
extern "C" void kernel_launch(void* const* d_in, const int* in_sizes, int n_in,
                              void* d_out, int out_size, void* d_ws, size_t ws_size,
                              hipStream_t stream) {
    const float* t    = (const float*)d_in[0];
    const float* x    = (const float*)d_in[1];
    const float* Wih0 = (const float*)d_in[2];
    const float* Whh0 = (const float*)d_in[3];
    const float* bih0 = (const float*)d_in[4];
    const float* bhh0 = (const float*)d_in[5];
    const float* Wih1 = (const float*)d_in[6];
    const float* Whh1 = (const float*)d_in[7];
    const float* bih1 = (const float*)d_in[8];
    const float* bhh1 = (const float*)d_in[9];
    float* out = (float*)d_out;

    float* imp = (float*)d_ws;                              // [B][S][36]
    float* h0  = imp + (size_t)BATCH * SEQ * KP0;           // [B][S][128]

    impute_kernel<<<BATCH, 32, 0, stream>>>(t, x, imp);

    const size_t smem = (size_t)(3 * HID + 48) * WSTR * sizeof(float);
    gru_wmma<KP0, KR0, 9, true><<<BATCH / 16, 256, smem, stream>>>(
        imp, Wih0, bih0, Whh0, bhh0, h0, nullptr);
    gru_wmma<KP1, KP1, 32, false><<<BATCH / 16, 256, smem, stream>>>(
        h0, Wih1, bih1, Whh1, bhh1, nullptr, out);
}